// AtomicConv_22754736734663
// MI455X (gfx1250) — compile-verified
//
#include <hip/hip_runtime.h>
#include <hip/hip_bf16.h>
#include <stdint.h>

// AtomicConv (ACNN) message passing, faithful to the reference's flat
// (K,E,1)->(E,K) reshape:
//   he[e,k] = rbf(params[(16e+k)/E], distances[(16e+k)%E])
//   out[dst[e], t, k] += he[e,k]   where t = one-hot index of feat[src[e]]
//
// A block-tile of 256 consecutive edges consumes a contiguous 4096-float
// window of `distances` -> staged into LDS with CDNA5 async copies
// (global_load_async_to_lds_b128 + s_wait_asynccnt), per-wave regions so no
// barrier is required.

#define KR   16
#define NT   8
#define TK   (KR * NT)
#define TILE 256

__device__ __forceinline__ void atom_add_f32(float* p, float v) {
#if __has_builtin(__builtin_amdgcn_global_atomic_fadd_f32)
  __builtin_amdgcn_global_atomic_fadd_f32(p, v);   // global_atomic_add_f32 (no return)
#else
  __hip_atomic_fetch_add(p, v, __ATOMIC_RELAXED, __HIP_MEMORY_SCOPE_AGENT);
#endif
}

// CDNA5 async global->LDS copy: per active lane, copies 16B to the LDS byte address lds_byte_addr
// from the 64-bit global address. Tracked by ASYNCcnt.
__device__ __forceinline__ void async_copy_b128_to_lds(uint32_t lds_byte_addr,
                                                       const void* gptr) {
  asm volatile("global_load_async_to_lds_b128 %0, %1, off"
               :
               : "v"(lds_byte_addr), "v"((uint64_t)(uintptr_t)gptr)
               : "memory");
}

__device__ __forceinline__ void wait_asynccnt0() {
  asm volatile("s_wait_asynccnt 0x0" ::: "memory");
}

__global__ __launch_bounds__(256) void acnn_zero_kernel(float4* __restrict__ out4, int n4) {
  int i = blockIdx.x * blockDim.x + threadIdx.x;
  if (i < n4) out4[i] = make_float4(0.f, 0.f, 0.f, 0.f);
}

__global__ __launch_bounds__(TILE) void acnn_edge_kernel(
    const float* __restrict__ feat,       // (N,1) atomic numbers
    const float* __restrict__ dist,       // (E,1)
    const float* __restrict__ rp,         // (16,3) [cutoff, mean, scaling]
    const float* __restrict__ f2u,        // (8,)
    const int*   __restrict__ src,        // (E,)
    const int*   __restrict__ dst,        // (E,)
    float*       __restrict__ out,        // (N,128)
    int nEdges) {
  __shared__ float s_cut[KR];
  __shared__ float s_mu[KR];
  __shared__ float s_sc[KR];
  __shared__ float s_pic[KR];
  __shared__ float s_dist[KR * TILE];     // 16KB staged distance window

  const int tid = threadIdx.x;
  if (tid < KR) {
    float c = rp[3 * tid + 0];
    s_cut[tid] = c;
    s_mu[tid]  = rp[3 * tid + 1];
    s_sc[tid]  = rp[3 * tid + 2];
    s_pic[tid] = 3.14159265358979323846f / c;
  }
  // features_to_use: uniform loads -> scalar registers
  float ft[NT];
#pragma unroll
  for (int j = 0; j < NT; ++j) ft[j] = f2u[j];
  __syncthreads();

  const uint32_t ldsbase = (uint32_t)(uintptr_t)(void*)&s_dist[0];
  const int wave = tid >> 5;
  const int lane = tid & 31;
  const int gstride = gridDim.x * TILE;

  for (int tile0 = blockIdx.x * TILE; tile0 < nEdges; tile0 += gstride) {
    // ---- stage this tile's distance window into LDS (per-wave 2KB region) ----
    // window float j holds dist[(gbase + j) mod nE]; chunks are 16B and never
    // straddle the wrap because gbase%4==0 and nE%4==0.
    const int gbase = (int)((16LL * (long long)tile0) % (long long)nEdges);
#pragma unroll
    for (int j = 0; j < 4; ++j) {
      const int f   = 512 * wave + (j * 32 + lane) * 4;  // float idx in window
      int idx = gbase + f;
      if (idx >= nEdges) idx -= nEdges;
      async_copy_b128_to_lds(ldsbase + (uint32_t)f * 4u, dist + idx);
    }

    // ---- overlap: gather src type while the async copy is in flight ----
    const int e = tile0 + tid;
    int t = -1;
    if (e < nEdges) {
      const float fv = feat[src[e]];
#pragma unroll
      for (int j = 0; j < NT; ++j) t = (fv == ft[j]) ? j : t;
    }

    // prefetch next tile's index arrays (global_prefetch_b8)
    if (tile0 + gstride < nEdges) {
      __builtin_prefetch(src + tile0 + gstride + tid, 0, 1);
      __builtin_prefetch(dst + tile0 + gstride + tid, 0, 1);
    }

    wait_asynccnt0();   // wave's own region complete; no cross-wave sharing

    if (e < nEdges && t >= 0) {
      // flat = 16e+k; q = flat / nE; distance comes from LDS window at 16*tid+k
      const int flat0 = KR * e;
      const int q0    = flat0 / nEdges;
      const int rem0  = flat0 - q0 * nEdges;
      int q1 = q0 + 1;
      if (q1 >= KR) q1 = KR - 1;   // q1 only used when a crossing actually occurs

      const float mu0 = s_mu[q0], sc0 = s_sc[q0], cu0 = s_cut[q0], pc0 = s_pic[q0];
      const float mu1 = s_mu[q1], sc1 = s_sc[q1], cu1 = s_cut[q1], pc1 = s_pic[q1];

      float dk[KR];
      const float4* sd = (const float4*)&s_dist[KR * tid];
#pragma unroll
      for (int j = 0; j < 4; ++j) {   // 4x ds_load_b128
        const float4 v = sd[j];
        dk[4 * j + 0] = v.x; dk[4 * j + 1] = v.y;
        dk[4 * j + 2] = v.z; dk[4 * j + 3] = v.w;
      }

      float* base = out + (size_t)dst[e] * TK + (size_t)t * KR;
#pragma unroll
      for (int k = 0; k < KR; ++k) {
        const bool  c  = (rem0 + k) >= nEdges;    // rare wrap within the row
        const float mu = c ? mu1 : mu0;
        const float sc = c ? sc1 : sc0;
        const float cu = c ? cu1 : cu0;
        const float pc = c ? pc1 : pc0;
        const float d  = dk[k];
        const float x  = d - mu;
        const float g  = __expf(-sc * x * x);                 // v_exp_f32
        const float cv = 0.5f * (__cosf(pc * d) + 1.0f);      // v_cos_f32
        const float w  = (d <= cu) ? g * cv : 0.0f;
        atom_add_f32(base + k, w);
      }
    }
    // Next iteration's async writes hit this wave's own region only after the
    // dk[] values were consumed (compiler inserts the DScnt wait), so there is
    // no LDS WAR hazard and no barrier in the loop.
  }
}

extern "C" void kernel_launch(void* const* d_in, const int* in_sizes, int n_in,
                              void* d_out, int out_size, void* d_ws, size_t ws_size,
                              hipStream_t stream) {
  const float* feat = (const float*)d_in[0];   // (N,1)
  const float* dist = (const float*)d_in[1];   // (E,1)
  const float* rp   = (const float*)d_in[2];   // (16,3)
  const float* f2u  = (const float*)d_in[3];   // (8,)
  const int*   src  = (const int*)d_in[4];     // (E,)
  const int*   dst  = (const int*)d_in[5];     // (E,)
  float*       out  = (float*)d_out;           // (N,128)

  const int nEdges = in_sizes[1];

  // Zero the output every call (timed graph replays would accumulate otherwise).
  const int n4 = out_size / 4;
  acnn_zero_kernel<<<(n4 + 255) / 256, 256, 0, stream>>>((float4*)out, n4);

  const int blocks = 1024;   // grid-stride over 256-edge tiles
  acnn_edge_kernel<<<blocks, TILE, 0, stream>>>(feat, dist, rp, f2u, src, dst, out, nEdges);

  (void)d_ws; (void)ws_size; (void)n_in;
}